// FlowGNN_90254442758602
// MI455X (gfx1250) — compile-verified
//
#include <hip/hip_runtime.h>
#include <math.h>

typedef __attribute__((ext_vector_type(2))) float v2f;
typedef __attribute__((ext_vector_type(8))) float v8f;

#define TMOD_W 518

__device__ __forceinline__ float silu_f(float v) {
    return v / (1.0f + __expf(-v));
}

// ---------------- zero fill ----------------
__global__ void fg_zero_kernel(float* __restrict__ p, size_t n) {
    size_t i = (size_t)blockIdx.x * blockDim.x + threadIdx.x;
    size_t stride = (size_t)gridDim.x * blockDim.x;
    for (; i < n; i += stride) p[i] = 0.0f;
}

// ---------------- time MLP: t -> t_mod (B x 518) ----------------
__global__ __launch_bounds__(128) void fg_time_mlp_kernel(
    const float* __restrict__ t, const float* __restrict__ Wt1,
    const float* __restrict__ bt1, const float* __restrict__ Wt2,
    const float* __restrict__ bt2, float* __restrict__ tmod) {
    __shared__ float hm[128];
    int b = blockIdx.x;
    int j = threadIdx.x;
    float tv = t[b];
    hm[j] = silu_f(tv * Wt1[j] + bt1[j]);
    __syncthreads();
    for (int k = j; k < TMOD_W; k += 128) {
        float acc = bt2[k];
        const float* w = Wt2 + (size_t)k * 128;
        #pragma unroll 8
        for (int q = 0; q < 128; ++q) acc += hm[q] * w[q];
        tmod[(size_t)b * TMOD_W + k] = acc;
    }
}

// ---------------- degree count (float) ----------------
__global__ void fg_count_kernel(const int* __restrict__ dst, float* __restrict__ cnt, int E) {
    int e = blockIdx.x * blockDim.x + threadIdx.x;
    if (e < E) unsafeAtomicAdd(cnt + dst[e], 1.0f);
}

__global__ void fg_invcnt_kernel(float* __restrict__ cnt, int N) {
    int i = blockIdx.x * blockDim.x + threadIdx.x;
    if (i < N) cnt[i] = 1.0f / fmaxf(cnt[i], 1.0f);
}

// ---------------- scatter, d=3 (layer 0 aggregation of x) ----------------
__global__ void fg_scatter3_kernel(const int* __restrict__ src, const int* __restrict__ dst,
                                   const float* __restrict__ x, float* __restrict__ agg3, int E) {
    int e = blockIdx.x * blockDim.x + threadIdx.x;
    if (e >= E) return;
    int s = src[e], d = dst[e];
    float a0 = x[(size_t)s * 3 + 0];
    float a1 = x[(size_t)s * 3 + 1];
    float a2 = x[(size_t)s * 3 + 2];
    unsafeAtomicAdd(agg3 + (size_t)d * 3 + 0, a0);
    unsafeAtomicAdd(agg3 + (size_t)d * 3 + 1, a1);
    unsafeAtomicAdd(agg3 + (size_t)d * 3 + 2, a2);
}

// ---------------- scatter, d=128: wave per edge, float4 per lane ----------------
__global__ __launch_bounds__(256) void fg_scatter128_kernel(
    const int* __restrict__ src, const int* __restrict__ dst,
    const float* __restrict__ h, float* __restrict__ agg, int E) {
    int wave = threadIdx.x >> 5;
    int lane = threadIdx.x & 31;
    int e = blockIdx.x * 8 + wave;
    if (e >= E) return;
    int s = src[e], d = dst[e];
    const float4 v = *(const float4*)(h + (size_t)s * 128 + lane * 4);
    float* p = agg + (size_t)d * 128 + lane * 4;
    unsafeAtomicAdd(p + 0, v.x);
    unsafeAtomicAdd(p + 1, v.y);
    unsafeAtomicAdd(p + 2, v.z);
    unsafeAtomicAdd(p + 3, v.w);
}

// ---------------- layer 0: K=3 GEMM + FiLM + SiLU (scalar) ----------------
__global__ __launch_bounds__(256) void fg_layer0_kernel(
    const float* __restrict__ agg3, const float* __restrict__ inv,
    const float* __restrict__ x,
    const float* __restrict__ Wl, const float* __restrict__ bl,
    const float* __restrict__ Wr,
    const float* __restrict__ tmod, const int* __restrict__ batch,
    float* __restrict__ out, int N) {
    size_t gid = (size_t)blockIdx.x * blockDim.x + threadIdx.x;
    if (gid >= (size_t)N * 128) return;
    int node = (int)(gid >> 7);
    int col = (int)(gid & 127);
    float iv = inv[node];
    float a0 = agg3[(size_t)node * 3 + 0] * iv;
    float a1 = agg3[(size_t)node * 3 + 1] * iv;
    float a2 = agg3[(size_t)node * 3 + 2] * iv;
    float x0 = x[(size_t)node * 3 + 0];
    float x1 = x[(size_t)node * 3 + 1];
    float x2 = x[(size_t)node * 3 + 2];
    float v = bl[col];
    v += a0 * Wl[col * 3 + 0] + a1 * Wl[col * 3 + 1] + a2 * Wl[col * 3 + 2];
    v += x0 * Wr[col * 3 + 0] + x1 * Wr[col * 3 + 1] + x2 * Wr[col * 3 + 2];
    int b = batch[node];
    float g  = tmod[(size_t)b * TMOD_W + col];         // offs[0]=0, gamma
    float be = tmod[(size_t)b * TMOD_W + 128 + col];   // beta
    v = v * (1.0f + g) + be;
    out[gid] = silu_f(v);
}

// ---------------- layer 1: 128->128 dual GEMM via V_WMMA_F32_16X16X4_F32 ----------------
// block: 256 threads = 8 waves; block tile = 16 nodes x 128 cols; wave w -> cols [16w,16w+16)
__global__ __launch_bounds__(256) void fg_layer1_wmma_kernel(
    const float* __restrict__ agg, const float* __restrict__ inv,
    const float* __restrict__ h,
    const float* __restrict__ Wl, const float* __restrict__ bl,
    const float* __restrict__ Wr,
    const float* __restrict__ tmod, const int* __restrict__ batch,
    float* __restrict__ out, int N) {
    __shared__ float sA[16][132];   // scaled aggregation tile (pad 132 -> conflict-free column reads)
    __shared__ float sH[16][132];   // h tile
    int tile_base = blockIdx.x * 16;

    // cooperative stage: 16x128 floats each, coalesced
    for (int i = threadIdx.x; i < 16 * 128; i += 256) {
        int r = i >> 7;
        int c = i & 127;
        int node = tile_base + r;
        float iv = inv[node];
        sA[r][c] = agg[(size_t)node * 128 + c] * iv;
        sH[r][c] = h[(size_t)node * 128 + c];
    }
    __syncthreads();

    int wave = threadIdx.x >> 5;
    int lane = threadIdx.x & 31;
    int row = lane & 15;            // A: M index
    int half = (lane >> 4) << 1;    // K sub-offset: 0 for lanes 0-15, 2 for lanes 16-31
    int col_base = wave * 16;
    int coll = lane & 15;           // B/D: N index
    int wrow = col_base + coll;     // weight row (output feature)

    v8f acc = {0.f, 0.f, 0.f, 0.f, 0.f, 0.f, 0.f, 0.f};

    // agg @ Wl.T
    #pragma unroll 4
    for (int kk = 0; kk < 128; kk += 4) {
        v2f a, bm;
        a.x = sA[row][kk + half];
        a.y = sA[row][kk + half + 1];
        bm.x = Wl[(size_t)wrow * 128 + kk + half];       // B[k][n] = Wl[n][k]
        bm.y = Wl[(size_t)wrow * 128 + kk + half + 1];
        acc = __builtin_amdgcn_wmma_f32_16x16x4_f32(false, a, false, bm,
                                                    (short)0, acc, false, false);
    }
    // + h @ Wr.T
    #pragma unroll 4
    for (int kk = 0; kk < 128; kk += 4) {
        v2f a, bm;
        a.x = sH[row][kk + half];
        a.y = sH[row][kk + half + 1];
        bm.x = Wr[(size_t)wrow * 128 + kk + half];
        bm.y = Wr[(size_t)wrow * 128 + kk + half + 1];
        acc = __builtin_amdgcn_wmma_f32_16x16x4_f32(false, a, false, bm,
                                                    (short)0, acc, false, false);
    }

    // epilogue: bias + FiLM(layer1: offs=256) + SiLU
    int colg = col_base + coll;
    float bias = bl[colg];
    int rhi = (lane >> 4) << 3;     // lanes 16-31 hold M = r+8
    #pragma unroll
    for (int r = 0; r < 8; ++r) {
        int node = tile_base + r + rhi;
        int b = batch[node];
        float v = acc[r] + bias;
        float g  = tmod[(size_t)b * TMOD_W + 256 + colg];
        float be = tmod[(size_t)b * TMOD_W + 384 + colg];
        v = v * (1.0f + g) + be;
        out[(size_t)node * 128 + colg] = silu_f(v);
    }
}

// ---------------- layer 2: 128->3 GEMM + FiLM (wave per node, shuffle reduce) ----------------
__global__ __launch_bounds__(256) void fg_layer2_kernel(
    const float* __restrict__ agg, const float* __restrict__ inv,
    const float* __restrict__ h,
    const float* __restrict__ Wl, const float* __restrict__ bl,
    const float* __restrict__ Wr,
    const float* __restrict__ tmod, const int* __restrict__ batch,
    float* __restrict__ out, int N) {
    int wave = threadIdx.x >> 5;
    int lane = threadIdx.x & 31;
    int node = blockIdx.x * 8 + wave;
    if (node >= N) return;
    const float4 av = *(const float4*)(agg + (size_t)node * 128 + lane * 4);
    const float4 hv = *(const float4*)(h + (size_t)node * 128 + lane * 4);
    float pa[3], ph[3];
    #pragma unroll
    for (int c = 0; c < 3; ++c) {
        const float* wl = Wl + (size_t)c * 128 + lane * 4;
        const float* wr = Wr + (size_t)c * 128 + lane * 4;
        pa[c] = av.x * wl[0] + av.y * wl[1] + av.z * wl[2] + av.w * wl[3];
        ph[c] = hv.x * wr[0] + hv.y * wr[1] + hv.z * wr[2] + hv.w * wr[3];
    }
    #pragma unroll
    for (int off = 16; off >= 1; off >>= 1) {
        #pragma unroll
        for (int c = 0; c < 3; ++c) {
            pa[c] += __shfl_xor(pa[c], off, 32);
            ph[c] += __shfl_xor(ph[c], off, 32);
        }
    }
    if (lane == 0) {
        float iv = inv[node];
        int b = batch[node];
        #pragma unroll
        for (int c = 0; c < 3; ++c) {
            float v = pa[c] * iv + bl[c] + ph[c];
            float g  = tmod[(size_t)b * TMOD_W + 512 + c];
            float be = tmod[(size_t)b * TMOD_W + 515 + c];
            out[(size_t)node * 3 + c] = v * (1.0f + g) + be;   // no SiLU on last layer
        }
    }
}

extern "C" void kernel_launch(void* const* d_in, const int* in_sizes, int n_in,
                              void* d_out, int out_size, void* d_ws, size_t ws_size,
                              hipStream_t stream) {
    const float* x   = (const float*)d_in[0];
    const float* t   = (const float*)d_in[1];
    const float* Wl0 = (const float*)d_in[2];
    const float* bl0 = (const float*)d_in[3];
    const float* Wr0 = (const float*)d_in[4];
    const float* Wl1 = (const float*)d_in[5];
    const float* bl1 = (const float*)d_in[6];
    const float* Wr1 = (const float*)d_in[7];
    const float* Wl2 = (const float*)d_in[8];
    const float* bl2 = (const float*)d_in[9];
    const float* Wr2 = (const float*)d_in[10];
    const float* Wt1 = (const float*)d_in[11];
    const float* bt1 = (const float*)d_in[12];
    const float* Wt2 = (const float*)d_in[13];
    const float* bt2 = (const float*)d_in[14];
    const int* edge_index  = (const int*)d_in[15];
    const int* batch_index = (const int*)d_in[16];
    float* out = (float*)d_out;

    const int N = in_sizes[0] / 3;
    const int E = in_sizes[15] / 2;
    const int B = in_sizes[1];
    const int* esrc = edge_index;
    const int* edst = edge_index + E;

    // workspace layout (floats)
    float* w = (float*)d_ws;
    size_t o = 0;
    float* tmod = w + o; o += ((size_t)B * TMOD_W + 255) & ~(size_t)255;
    float* inv  = w + o; o += (size_t)N;
    float* agg3 = w + o; o += (size_t)N * 3;
    float* h1   = w + o; o += (size_t)N * 128;
    float* agg  = w + o; o += (size_t)N * 128;   // reused for layer1 and layer2 aggregation
    float* h2   = w + o; o += (size_t)N * 128;
    (void)ws_size; (void)n_in; (void)out_size;

    const int ZB = 2048;  // blocks for zero-fill grid-stride

    // degree -> 1/max(deg,1)
    fg_zero_kernel<<<ZB, 256, 0, stream>>>(inv, (size_t)N);
    fg_count_kernel<<<(E + 255) / 256, 256, 0, stream>>>(edst, inv, E);
    fg_invcnt_kernel<<<(N + 255) / 256, 256, 0, stream>>>(inv, N);

    // time modulation table
    fg_time_mlp_kernel<<<B, 128, 0, stream>>>(t, Wt1, bt1, Wt2, bt2, tmod);

    // ---- layer 0 ----
    fg_zero_kernel<<<ZB, 256, 0, stream>>>(agg3, (size_t)N * 3);
    fg_scatter3_kernel<<<(E + 255) / 256, 256, 0, stream>>>(esrc, edst, x, agg3, E);
    fg_layer0_kernel<<<(int)(((size_t)N * 128 + 255) / 256), 256, 0, stream>>>(
        agg3, inv, x, Wl0, bl0, Wr0, tmod, batch_index, h1, N);

    // ---- layer 1 (WMMA) ----
    fg_zero_kernel<<<ZB, 256, 0, stream>>>(agg, (size_t)N * 128);
    fg_scatter128_kernel<<<(E + 7) / 8, 256, 0, stream>>>(esrc, edst, h1, agg, E);
    fg_layer1_wmma_kernel<<<N / 16, 256, 0, stream>>>(
        agg, inv, h1, Wl1, bl1, Wr1, tmod, batch_index, h2, N);

    // ---- layer 2 ----
    fg_zero_kernel<<<ZB, 256, 0, stream>>>(agg, (size_t)N * 128);
    fg_scatter128_kernel<<<(E + 7) / 8, 256, 0, stream>>>(esrc, edst, h2, agg, E);
    fg_layer2_kernel<<<(N + 7) / 8, 256, 0, stream>>>(
        agg, inv, h2, Wl2, bl2, Wr2, tmod, batch_index, out, N);
}